// VectorDiscoBlock_79250736546297
// MI455X (gfx1250) — compile-verified
//
#include <hip/hip_runtime.h>
#include <cmath>

typedef float v2f __attribute__((ext_vector_type(2)));
typedef float v8f __attribute__((ext_vector_type(8)));
typedef unsigned int u32x4 __attribute__((ext_vector_type(4)));
typedef int i32x8 __attribute__((ext_vector_type(8)));
typedef int i32x4 __attribute__((ext_vector_type(4)));

#define PTS 65160   // 181*360
#define NSC 32
#define NVC 16
#define NNB 16
#define TILE 16
#define SUB 8

// LDS partition (float offsets), total 24192 floats = 94.5 KB dynamic LDS
#define OFF_XSG 0        // [8][516]  gathered scalar feats  (sub-tile)
#define OFF_XVG 4128     // [8][516]  gathered vector feats
#define OFF_ZS  8256     // [16][196] z for scalar-output GEMM (ik=192)
#define OFF_ZV0 11392    // [16][196] z for vector-output GEMM, c=0
#define OFF_ZV1 14528    // [16][196] z for vector-output GEMM, c=1
#define OFF_PSS 17664    // [4][8][16]      psi_ss sub-tile
#define OFF_PSV 18176    // [4][8][16][2]   psi_sv
#define OFF_PVS 19200    // [4][8][16][2]   psi_vs
#define OFF_PVV 20224    // [4][8][16][2][2] psi_vv
#define OFF_SB  22272    // [16][36]  s tile staging (B-fragment layout)
#define OFF_HB  22848    // [16][68]  hidden tile staging
#define OFF_IDX 23936    // [16][16]  idx tile (ints)
#define LDS_FLOATS 24192

__device__ __forceinline__ float gelu_exact(float x) {
  return 0.5f * x * (1.0f + erff(x * 0.70710678118654752440f));
}

// ---------------------------------------------------------------------------
// Tensor Data Mover: 2D tile load (global -> LDS), fp32 elements.
//   tile = tile_d0 contiguous elems  x  n_rows rows, row stride = stride_e
//   tensor_dim0 = valid_d0 (hardware returns zeros beyond -> handles grid tail)
// D# layout per CDNA5 ISA (group0 128b, group1 256b).
// clang-23 toolchain: 6-arg builtin (g0, g1, g2, g3, extra, cpol).
// ---------------------------------------------------------------------------
__device__ __forceinline__ void tdm_load_2d(const float* gptr, unsigned lds_byte,
                                            unsigned tile_d0, unsigned valid_d0,
                                            unsigned n_rows,
                                            unsigned long long stride_e) {
  unsigned long long ga = (unsigned long long)(uintptr_t)gptr;
  u32x4 g0;
  g0[0] = 1u;                                            // count=1, user desc
  g0[1] = lds_byte;                                      // lds_addr (bytes)
  g0[2] = (unsigned)(ga & 0xffffffffu);                  // global_addr[31:0]
  g0[3] = (unsigned)((ga >> 32) & 0x01ffffffu) | (2u << 30);  // [56:32] | type=2
  i32x8 g1;
  g1[0] = (int)(2u << 16);                               // data_size=2 (4B)
  g1[1] = (int)((valid_d0 & 0xffffu) << 16);             // tensor_dim0[15:0]
  g1[2] = (int)((valid_d0 >> 16) | ((n_rows & 0xffffu) << 16)); // d0 hi | tensor_dim1 lo
  g1[3] = (int)((tile_d0 & 0xffffu) << 16);              // tensor_dim1 hi=0 | tile_dim0
  g1[4] = (int)(n_rows & 0xffffu);                       // tile_dim1 | tile_dim2=0
  g1[5] = (int)(unsigned)(stride_e & 0xffffffffu);       // tensor_dim0_stride lo
  g1[6] = (int)(unsigned)((stride_e >> 32) & 0xffffu);   // stride hi | dim1_stride=0
  g1[7] = 0;
  i32x4 z4 = {0, 0, 0, 0};
  i32x8 z8 = {0, 0, 0, 0, 0, 0, 0, 0};
  __builtin_amdgcn_tensor_load_to_lds(g0, g1, z4, z4, z8, 0);
}

// ---------------------------------------------------------------------------
// Kernel A: pack channel-major fields into point-major rows of 32 floats so
// each neighbor gather in the main kernel is one contiguous 128B cacheline.
// xsT[p][i] = xs[i][p];   xvT[p][2i+c] = xv[i][p][c]
// ---------------------------------------------------------------------------
__global__ __launch_bounds__(256) void pack_kernel(const float* __restrict__ xs,
                                                   const float* __restrict__ xv,
                                                   float* __restrict__ xsT,
                                                   float* __restrict__ xvT) {
  __shared__ float t[32][65];
  const int p0 = blockIdx.x * 64;
  const int pl = threadIdx.x & 63;
  const int cg = threadIdx.x >> 6;          // 0..3
  const int p  = min(p0 + pl, PTS - 1);

  #pragma unroll
  for (int it = 0; it < 8; ++it) {
    int i = it * 4 + cg;
    t[i][pl] = xs[i * PTS + p];             // coalesced over pl
  }
  __syncthreads();
  for (int e = threadIdx.x; e < 64 * 32; e += 256) {
    int pp = e >> 5, ii = e & 31;
    if (p0 + pp < PTS) xsT[(p0 + pp) * 32 + ii] = t[ii][pp];  // coalesced
  }
  __syncthreads();

  const float2* xv2 = (const float2*)xv;
  #pragma unroll
  for (int it = 0; it < 4; ++it) {
    int i = it * 4 + cg;
    float2 v = xv2[i * PTS + p];
    t[2 * i + 0][pl] = v.x;
    t[2 * i + 1][pl] = v.y;
  }
  __syncthreads();
  for (int e = threadIdx.x; e < 64 * 32; e += 256) {
    int pp = e >> 5, ii = e & 31;
    if (p0 + pp < PTS) xvT[(p0 + pp) * 32 + ii] = t[ii][pp];
  }
}

// ---------------------------------------------------------------------------
// Main kernel: one wave32 per 16-point tile. TDM streams psi into LDS, VALU
// psi-contraction feeds fp32 WMMA GEMMs for all W / MLP / gate contractions.
// ---------------------------------------------------------------------------
__global__ __launch_bounds__(32) void disco_kernel(
    const float* __restrict__ xs, const float* __restrict__ xv,
    const int* __restrict__ idx,
    const float* __restrict__ psi_ss, const float* __restrict__ psi_sv,
    const float* __restrict__ psi_vs, const float* __restrict__ psi_vv,
    const float* __restrict__ W_ss, const float* __restrict__ W_vs,
    const float* __restrict__ W_sv, const float* __restrict__ W_vv,
    const float* __restrict__ bias_s,
    const float* __restrict__ mlp_w1, const float* __restrict__ mlp_b1,
    const float* __restrict__ mlp_w2, const float* __restrict__ mlp_b2,
    const float* __restrict__ gate_w, const float* __restrict__ gate_b,
    const float* __restrict__ xsT, const float* __restrict__ xvT,
    float* __restrict__ out_s, float* __restrict__ out_v)
{
  extern __shared__ float sm[];
  int* sIdx = (int*)&sm[OFF_IDX];
  const int lane = threadIdx.x;
  const int h    = lane >> 4;     // half-wave (selects K pair / M offset 8)
  const int l15  = lane & 15;
  const int p0   = blockIdx.x * TILE;
  const unsigned ldsb = (unsigned)__builtin_amdgcn_groupstaticsize();

  // prefetch streamed psi for this tile into GL2
  __builtin_prefetch(psi_ss + (size_t)p0 * NNB, 0, 1);
  __builtin_prefetch(psi_vv + (size_t)p0 * NNB * 4, 0, 1);

  // idx tile
  for (int t = lane; t < TILE * NNB; t += 32) {
    int j = t >> 4, n = t & 15;
    int pj = min(p0 + j, PTS - 1);
    sIdx[t] = idx[pj * NNB + n];
  }

  for (int sub = 0; sub < 2; ++sub) {
    const int j0 = sub * SUB;
    const unsigned pb = (unsigned)(p0 + j0);          // <= PTS
    const unsigned rem = (unsigned)PTS - pb;          // valid points in dim0

    // drain prior-subtile LDS reads before TDM overwrites psi regions
    asm volatile("s_wait_dscnt 0" ::: "memory");

    // ---- psi sub-tiles via Tensor Data Mover (global -> LDS DMA) ----
    tdm_load_2d(psi_ss + (size_t)pb * 16, ldsb + OFF_PSS * 4,
                SUB * 16, rem * 16, 4, (unsigned long long)PTS * 16);
    tdm_load_2d(psi_sv + (size_t)pb * 32, ldsb + OFF_PSV * 4,
                SUB * 32, rem * 32, 4, (unsigned long long)PTS * 32);
    tdm_load_2d(psi_vs + (size_t)pb * 32, ldsb + OFF_PVS * 4,
                SUB * 32, rem * 32, 4, (unsigned long long)PTS * 32);
    tdm_load_2d(psi_vv + (size_t)pb * 64, ldsb + OFF_PVV * 4,
                SUB * 64, rem * 64, 4, (unsigned long long)PTS * 64);

    // ---- neighbor gather (overlaps TDM): 128B row per (point, neighbor) ----
    for (int rr = 0; rr < SUB * NNB; rr += 4) {
      int row = rr + (lane >> 3);
      int c4  = (lane & 7) * 4;
      int j = row >> 4, n = row & 15;
      int q = sIdx[(j0 + j) * NNB + n];
      float4 a = *(const float4*)(xsT + (size_t)q * 32 + c4);
      *(float4*)(&sm[OFF_XSG + j * 516 + n * 32 + c4]) = a;
      float4 b = *(const float4*)(xvT + (size_t)q * 32 + c4);
      *(float4*)(&sm[OFF_XVG + j * 516 + n * 32 + c4]) = b;
    }

    // psi tiles must be resident before the contraction
    __builtin_amdgcn_s_wait_tensorcnt(0);

    // ---- z scalar-input path: lane = scalar channel i (32 lanes) ----
    for (int j = 0; j < SUB; ++j) {
      float zss[4]  = {0, 0, 0, 0};
      float zsv0[4] = {0, 0, 0, 0};
      float zsv1[4] = {0, 0, 0, 0};
      #pragma unroll
      for (int n = 0; n < NNB; ++n) {
        float x = sm[OFF_XSG + j * 516 + n * 32 + lane];
        #pragma unroll
        for (int k = 0; k < 4; ++k) {
          zss[k]  = fmaf(sm[OFF_PSS + (k * 8 + j) * 16 + n], x, zss[k]);
          zsv0[k] = fmaf(sm[OFF_PSV + ((k * 8 + j) * 16 + n) * 2 + 0], x, zsv0[k]);
          zsv1[k] = fmaf(sm[OFF_PSV + ((k * 8 + j) * 16 + n) * 2 + 1], x, zsv1[k]);
        }
      }
      int jg = j0 + j;
      #pragma unroll
      for (int k = 0; k < 4; ++k) {
        sm[OFF_ZS  + jg * 196 + lane * 4 + k] = zss[k];   // ik = i*4+k (0..127)
        sm[OFF_ZV0 + jg * 196 + lane * 4 + k] = zsv0[k];
        sm[OFF_ZV1 + jg * 196 + lane * 4 + k] = zsv1[k];
      }
    }

    // ---- z vector-input path: half-wave handles alternating points, i=l15 ----
    for (int jp = 0; jp < 4; ++jp) {
      int jl = jp * 2 + h;
      float zvs[4]  = {0, 0, 0, 0};
      float zvv0[4] = {0, 0, 0, 0};
      float zvv1[4] = {0, 0, 0, 0};
      #pragma unroll
      for (int n = 0; n < NNB; ++n) {
        float x0 = sm[OFF_XVG + jl * 516 + n * 32 + l15 * 2 + 0];
        float x1 = sm[OFF_XVG + jl * 516 + n * 32 + l15 * 2 + 1];
        #pragma unroll
        for (int k = 0; k < 4; ++k) {
          int bvs = OFF_PVS + ((k * 8 + jl) * 16 + n) * 2;
          zvs[k]  = fmaf(sm[bvs + 0], x0, fmaf(sm[bvs + 1], x1, zvs[k]));
          int bvv = OFF_PVV + ((k * 8 + jl) * 16 + n) * 4;   // [c][d]
          zvv0[k] = fmaf(sm[bvv + 0], x0, fmaf(sm[bvv + 1], x1, zvv0[k]));
          zvv1[k] = fmaf(sm[bvv + 2], x0, fmaf(sm[bvv + 3], x1, zvv1[k]));
        }
      }
      int jg = j0 + jl;
      #pragma unroll
      for (int k = 0; k < 4; ++k) {
        sm[OFF_ZS  + jg * 196 + 128 + l15 * 4 + k] = zvs[k];  // ik = 128+i*4+k
        sm[OFF_ZV0 + jg * 196 + 128 + l15 * 4 + k] = zvv0[k];
        sm[OFF_ZV1 + jg * 196 + 128 + l15 * 4 + k] = zvv1[k];
      }
    }
  }

  // ======================= fp32 WMMA GEMM phase ============================
  // s(32x16) = [W_ss|W_vs](32x192) @ ZS(192x16) + bias
  // v_c(16x16) = [W_sv|W_vv](16x192) @ ZVc(192x16)
  v8f accS0 = {}, accS1 = {}, accV0 = {}, accV1 = {};
  #pragma unroll
  for (int r = 0; r < 8; ++r) {
    accS0[r] = bias_s[r + 8 * h];
    accS1[r] = bias_s[16 + r + 8 * h];
  }
  // ik < 128: scalar-input block (i = t)
  #pragma unroll 4
  for (int t = 0; t < 32; ++t) {
    const int col = 4 * t + 2 * h;
    v2f b   = *(const v2f*)(&sm[OFF_ZS  + l15 * 196 + col]);
    v2f b0  = *(const v2f*)(&sm[OFF_ZV0 + l15 * 196 + col]);
    v2f b1  = *(const v2f*)(&sm[OFF_ZV1 + l15 * 196 + col]);
    v2f a0  = *(const v2f*)(W_ss + l15 * 128 + col);
    v2f a1  = *(const v2f*)(W_ss + (16 + l15) * 128 + col);
    v2f avv = *(const v2f*)(W_sv + l15 * 128 + col);
    accS0 = __builtin_amdgcn_wmma_f32_16x16x4_f32(false, a0,  false, b,  (short)0, accS0, false, false);
    accS1 = __builtin_amdgcn_wmma_f32_16x16x4_f32(false, a1,  false, b,  (short)0, accS1, false, false);
    accV0 = __builtin_amdgcn_wmma_f32_16x16x4_f32(false, avv, false, b0, (short)0, accV0, false, false);
    accV1 = __builtin_amdgcn_wmma_f32_16x16x4_f32(false, avv, false, b1, (short)0, accV1, false, false);
  }
  // ik >= 128: vector-input block (i = t)
  #pragma unroll 4
  for (int t = 0; t < 16; ++t) {
    const int col  = 4 * t + 2 * h;
    const int colz = 128 + col;
    v2f b   = *(const v2f*)(&sm[OFF_ZS  + l15 * 196 + colz]);
    v2f b0  = *(const v2f*)(&sm[OFF_ZV0 + l15 * 196 + colz]);
    v2f b1  = *(const v2f*)(&sm[OFF_ZV1 + l15 * 196 + colz]);
    v2f a0  = *(const v2f*)(W_vs + l15 * 64 + col);
    v2f a1  = *(const v2f*)(W_vs + (16 + l15) * 64 + col);
    v2f avv = *(const v2f*)(W_vv + l15 * 64 + col);
    accS0 = __builtin_amdgcn_wmma_f32_16x16x4_f32(false, a0,  false, b,  (short)0, accS0, false, false);
    accS1 = __builtin_amdgcn_wmma_f32_16x16x4_f32(false, a1,  false, b,  (short)0, accS1, false, false);
    accV0 = __builtin_amdgcn_wmma_f32_16x16x4_f32(false, avv, false, b0, (short)0, accV0, false, false);
    accV1 = __builtin_amdgcn_wmma_f32_16x16x4_f32(false, avv, false, b1, (short)0, accV1, false, false);
  }

  // gelu(s) -> SB (B-fragment layout [j][ch])
  #pragma unroll
  for (int r = 0; r < 8; ++r) {
    sm[OFF_SB + l15 * 36 + (r + 8 * h)]      = gelu_exact(accS0[r]);
    sm[OFF_SB + l15 * 36 + (16 + r + 8 * h)] = gelu_exact(accS1[r]);
  }

  // h = gelu(w1(64x32) @ s + b1) -> HB
  for (int m2 = 0; m2 < 4; ++m2) {
    v8f accH = {};
    #pragma unroll
    for (int r = 0; r < 8; ++r) accH[r] = mlp_b1[16 * m2 + r + 8 * h];
    #pragma unroll
    for (int t = 0; t < 8; ++t) {
      int col = 4 * t + 2 * h;
      v2f a = *(const v2f*)(mlp_w1 + (16 * m2 + l15) * 32 + col);
      v2f b = *(const v2f*)(&sm[OFF_SB + l15 * 36 + col]);
      accH = __builtin_amdgcn_wmma_f32_16x16x4_f32(false, a, false, b, (short)0, accH, false, false);
    }
    #pragma unroll
    for (int r = 0; r < 8; ++r)
      sm[OFF_HB + l15 * 68 + (16 * m2 + r + 8 * h)] = gelu_exact(accH[r]);
  }

  // s2 = gelu_s + w2(32x64) @ h + b2   (C initialized from LDS gelu_s)
  v8f s20 = {}, s21 = {};
  #pragma unroll
  for (int r = 0; r < 8; ++r) {
    s20[r] = sm[OFF_SB + l15 * 36 + (r + 8 * h)]      + mlp_b2[r + 8 * h];
    s21[r] = sm[OFF_SB + l15 * 36 + (16 + r + 8 * h)] + mlp_b2[16 + r + 8 * h];
  }
  #pragma unroll
  for (int t = 0; t < 16; ++t) {
    int col = 4 * t + 2 * h;
    v2f b  = *(const v2f*)(&sm[OFF_HB + l15 * 68 + col]);
    v2f a0 = *(const v2f*)(mlp_w2 + l15 * 64 + col);
    v2f a1 = *(const v2f*)(mlp_w2 + (16 + l15) * 64 + col);
    s20 = __builtin_amdgcn_wmma_f32_16x16x4_f32(false, a0, false, b, (short)0, s20, false, false);
    s21 = __builtin_amdgcn_wmma_f32_16x16x4_f32(false, a1, false, b, (short)0, s21, false, false);
  }

  // stage s2 for the gate GEMM
  #pragma unroll
  for (int r = 0; r < 8; ++r) {
    sm[OFF_SB + l15 * 36 + (r + 8 * h)]      = s20[r];
    sm[OFF_SB + l15 * 36 + (16 + r + 8 * h)] = s21[r];
  }

  // gate = gate_w(16x32) @ s2 + gate_b
  v8f g = {};
  #pragma unroll
  for (int r = 0; r < 8; ++r) g[r] = gate_b[r + 8 * h];
  #pragma unroll
  for (int t = 0; t < 8; ++t) {
    int col = 4 * t + 2 * h;
    v2f a = *(const v2f*)(gate_w + l15 * 32 + col);
    v2f b = *(const v2f*)(&sm[OFF_SB + l15 * 36 + col]);
    g = __builtin_amdgcn_wmma_f32_16x16x4_f32(false, a, false, b, (short)0, g, false, false);
  }

  // ---- fused residual outputs ----
  const int p = p0 + l15;
  if (p < PTS) {
    #pragma unroll
    for (int r = 0; r < 8; ++r) {
      int o0 = r + 8 * h, o1 = 16 + r + 8 * h;
      out_s[o0 * PTS + p] = xs[o0 * PTS + p] + s20[r];
      out_s[o1 * PTS + p] = xs[o1 * PTS + p] + s21[r];
    }
    const float2* xv2 = (const float2*)xv;
    float2* ov2 = (float2*)out_v;
    #pragma unroll
    for (int r = 0; r < 8; ++r) {
      int o = r + 8 * h;
      float gm = 1.0f + g[r];                 // v + gate*v = v*(1+gate)
      float2 base = xv2[o * PTS + p];
      float2 res;
      res.x = base.x + accV0[r] * gm;
      res.y = base.y + accV1[r] * gm;
      ov2[o * PTS + p] = res;
    }
  }
}

extern "C" void kernel_launch(void* const* d_in, const int* in_sizes, int n_in,
                              void* d_out, int out_size, void* d_ws, size_t ws_size,
                              hipStream_t stream) {
  const float* xs     = (const float*)d_in[0];
  const float* xv     = (const float*)d_in[1];
  const int*   idx    = (const int*)d_in[2];
  const float* psi_ss = (const float*)d_in[3];
  const float* psi_sv = (const float*)d_in[4];
  const float* psi_vs = (const float*)d_in[5];
  const float* psi_vv = (const float*)d_in[6];
  const float* W_ss   = (const float*)d_in[7];
  const float* W_vs   = (const float*)d_in[8];
  const float* W_sv   = (const float*)d_in[9];
  const float* W_vv   = (const float*)d_in[10];
  const float* bias_s = (const float*)d_in[11];
  const float* mlp_w1 = (const float*)d_in[12];
  const float* mlp_b1 = (const float*)d_in[13];
  const float* mlp_w2 = (const float*)d_in[14];
  const float* mlp_b2 = (const float*)d_in[15];
  const float* gate_w = (const float*)d_in[16];
  const float* gate_b = (const float*)d_in[17];

  float* out_s = (float*)d_out;
  float* out_v = out_s + (size_t)NSC * PTS;
  float* xsT = (float*)d_ws;                 // [P][32]
  float* xvT = xsT + (size_t)32 * PTS;       // [P][32]

  pack_kernel<<<(PTS + 63) / 64, 256, 0, stream>>>(xs, xv, xsT, xvT);
  disco_kernel<<<(PTS + TILE - 1) / TILE, 32, LDS_FLOATS * 4, stream>>>(
      xs, xv, idx, psi_ss, psi_sv, psi_vs, psi_vv,
      W_ss, W_vs, W_sv, W_vv, bias_s,
      mlp_w1, mlp_b1, mlp_w2, mlp_b2, gate_w, gate_b,
      xsT, xvT, out_s, out_v);
}